// MyGCN2_66297115181468
// MI455X (gfx1250) — compile-verified
//
#include <hip/hip_runtime.h>
#include <hip/hip_bf16.h>

// ---------------------------------------------------------------------------
// 2-layer GCN on MI455X (gfx1250, wave32).
//   h1   = x @ W1                         (WMMA f32 16x16x4, reg-resident A+B)
//   h    = scatter(norm * h1[src]) + dinv^2*h1 + b1
//   h2   = h @ W2                         (tiny, scalar FMA)
//   out  = scatter(norm * h2[src]) + dinv^2*h2 + b2
// norm[e] = dinv[src]*dinv[dst], dinv = 1/sqrt(in_deg + 1 self-loop)
// Gather/scatter is L2-resident (h = 25.6MB << 192MB L2); GEMM keeps fp32.
// ---------------------------------------------------------------------------

#define NN      100000
#define NE      1600000
#define IN_F    128
#define HID     64
#define OUT_F   7

typedef __attribute__((ext_vector_type(2))) float v2f;
typedef __attribute__((ext_vector_type(8))) float v8f;

// ---------------- init: deg=1 (self loop), agg1=0, out=0 -------------------
__global__ __launch_bounds__(256) void k_init(float* __restrict__ deg,
                                              float* __restrict__ agg1,
                                              float* __restrict__ out) {
    int i = blockIdx.x * 256 + threadIdx.x;
    if (i < NN * HID)   agg1[i] = 0.0f;
    if (i < NN * OUT_F) out[i]  = 0.0f;
    if (i < NN)         deg[i]  = 1.0f;
}

// ---------------- degree accumulation over edges ---------------------------
__global__ __launch_bounds__(256) void k_degree(const int* __restrict__ dst,
                                                float* __restrict__ deg) {
    int e = blockIdx.x * 256 + threadIdx.x;
    if (e < NE) atomicAdd(&deg[dst[e]], 1.0f);
}

// ---------------- dinv = rsqrt(deg) in place -------------------------------
__global__ __launch_bounds__(256) void k_dinv(float* __restrict__ deg) {
    int i = blockIdx.x * 256 + threadIdx.x;
    if (i < NN) deg[i] = rsqrtf(deg[i]);   // deg >= 1 always (self loop)
}

// ---------------- GEMM1: h1[NN,64] = x[NN,128] @ W1[128,64] via WMMA -------
// Block = 256 threads = 8 waves; block computes 128 nodes. Wave w owns
// nodes [w*16, w*16+16) and iterates 4 N-tiles of 16. K=128 -> 32 wmma steps.
// A and B fragments are fully register-resident per tile so the 32 WMMAs run
// back-to-back (one ds-load clause + one wait per N-tile, not per step).
#define XPAD 132   // padded LDS row stride for x tile (bank spread)
__global__ __launch_bounds__(256) void k_gemm1(const float* __restrict__ x,
                                               const float* __restrict__ W1,
                                               float* __restrict__ h1) {
    extern __shared__ float smem[];
    float* sW = smem;              // 128*64  = 8192 floats (32KB)
    float* sX = smem + IN_F * HID; // 128*132 = 16896 floats (~66KB)

    const int tid = threadIdx.x;
    const int nodeBase = blockIdx.x * 128;

    for (int i = tid; i < IN_F * HID; i += 256) sW[i] = W1[i];
    for (int i = tid; i < 128 * IN_F; i += 256) {
        int r = i >> 7, c = i & 127;
        int node = nodeBase + r;
        sX[r * XPAD + c] = (node < NN) ? x[node * IN_F + c] : 0.0f;
    }
    __syncthreads();

    const int wave = tid >> 5;
    const int lane = tid & 31;
    const int g    = lane >> 4;   // half-wave: selects K pair
    const int mn   = lane & 15;   // M (for A) / N (for B,D)
    const int rowOff = (wave * 16 + mn) * XPAD;
    const bool full = (nodeBase + 128 <= NN);   // 781 of 782 blocks

    // A fragments: lane (g*16+m) holds A[m][4k+2g], A[m][4k+2g+1]
    v2f afrag[32];
    #pragma unroll
    for (int k4 = 0; k4 < 32; ++k4) {
        int k0 = k4 * 4 + 2 * g;
        afrag[k4][0] = sX[rowOff + k0];
        afrag[k4][1] = sX[rowOff + k0 + 1];
    }

    for (int t = 0; t < 4; ++t) {           // N tiles of 16 (HID=64)
        // B fragments: lane (g*16+n) holds B[4k+2g][n], B[4k+2g+1][n]
        v2f bfrag[32];
        #pragma unroll
        for (int k4 = 0; k4 < 32; ++k4) {
            int k0 = k4 * 4 + 2 * g;
            bfrag[k4][0] = sW[(k0    ) * HID + t * 16 + mn];
            bfrag[k4][1] = sW[(k0 + 1) * HID + t * 16 + mn];
        }
        v8f acc = {};
        #pragma unroll
        for (int k4 = 0; k4 < 32; ++k4) {
            acc = __builtin_amdgcn_wmma_f32_16x16x4_f32(
                /*neg_a=*/false, afrag[k4], /*neg_b=*/false, bfrag[k4],
                /*c_mod=*/(short)0, acc, /*reuse_a=*/false, /*reuse_b=*/false);
        }
        // D 16x16 f32: VGPR r -> M = r + 8*g, N = mn
        if (full) {
            #pragma unroll
            for (int r = 0; r < 8; ++r) {
                int node = nodeBase + wave * 16 + r + 8 * g;
                h1[node * HID + t * 16 + mn] = acc[r];
            }
        } else {
            #pragma unroll
            for (int r = 0; r < 8; ++r) {
                int node = nodeBase + wave * 16 + r + 8 * g;
                if (node < NN) h1[node * HID + t * 16 + mn] = acc[r];
            }
        }
    }
}

// ---------------- layer-1 edge scatter: one thread per (edge, feat) --------
__global__ __launch_bounds__(256) void k_scatter1(const int* __restrict__ ei,
                                                  const float* __restrict__ dinv,
                                                  const float* __restrict__ h1,
                                                  float* __restrict__ agg1) {
    int gid = blockIdx.x * 256 + threadIdx.x;  // NE*64 threads
    if (gid >= NE * HID) return;
    int e = gid >> 6;
    int f = gid & 63;
    int s = ei[e];
    int d = ei[NE + e];
    float w = dinv[s] * dinv[d];
    atomicAdd(&agg1[d * HID + f], w * h1[s * HID + f]);
}

// ---------------- finish1: h = agg1 + dinv^2*h1 + b1 (into agg1) -----------
__global__ __launch_bounds__(256) void k_finish1(const float* __restrict__ dinv,
                                                 const float* __restrict__ h1,
                                                 const float* __restrict__ b1,
                                                 float* __restrict__ agg1) {
    int gid = blockIdx.x * 256 + threadIdx.x;
    if (gid >= NN * HID) return;
    int node = gid >> 6;
    int f    = gid & 63;
    float dv = dinv[node];
    agg1[gid] = agg1[gid] + dv * dv * h1[gid] + b1[f];
}

// ---------------- GEMM2: h2[NN,7] = h[NN,64] @ W2[64,7] --------------------
__global__ __launch_bounds__(256) void k_gemm2(const float* __restrict__ h,
                                               const float* __restrict__ W2,
                                               float* __restrict__ h2) {
    int i = blockIdx.x * 256 + threadIdx.x;
    if (i >= NN) return;
    float acc[OUT_F];
    #pragma unroll
    for (int f = 0; f < OUT_F; ++f) acc[f] = 0.0f;
    const float* hr = h + i * HID;
    #pragma unroll 4
    for (int k = 0; k < HID; ++k) {
        float hv = hr[k];
        #pragma unroll
        for (int f = 0; f < OUT_F; ++f) acc[f] += hv * W2[k * OUT_F + f];
    }
    #pragma unroll
    for (int f = 0; f < OUT_F; ++f) h2[i * OUT_F + f] = acc[f];
}

// ---------------- layer-2 edge scatter: one thread per edge, 7 feats -------
__global__ __launch_bounds__(256) void k_scatter2(const int* __restrict__ ei,
                                                  const float* __restrict__ dinv,
                                                  const float* __restrict__ h2,
                                                  float* __restrict__ out) {
    int e = blockIdx.x * 256 + threadIdx.x;
    if (e >= NE) return;
    int s = ei[e];
    int d = ei[NE + e];
    float w = dinv[s] * dinv[d];
    #pragma unroll
    for (int f = 0; f < OUT_F; ++f)
        atomicAdd(&out[d * OUT_F + f], w * h2[s * OUT_F + f]);
}

// ---------------- finish2: out += dinv^2*h2 + b2 ---------------------------
__global__ __launch_bounds__(256) void k_finish2(const float* __restrict__ dinv,
                                                 const float* __restrict__ h2,
                                                 const float* __restrict__ b2,
                                                 float* __restrict__ out) {
    int gid = blockIdx.x * 256 + threadIdx.x;
    if (gid >= NN * OUT_F) return;
    int node = gid / OUT_F;
    int f    = gid - node * OUT_F;
    float dv = dinv[node];
    out[gid] += dv * dv * h2[gid] + b2[f];
}

extern "C" void kernel_launch(void* const* d_in, const int* in_sizes, int n_in,
                              void* d_out, int out_size, void* d_ws, size_t ws_size,
                              hipStream_t stream) {
    const float* x  = (const float*)d_in[0];
    const float* W1 = (const float*)d_in[1];
    const float* b1 = (const float*)d_in[2];
    const float* W2 = (const float*)d_in[3];
    const float* b2 = (const float*)d_in[4];
    const int*   ei = (const int*)d_in[5];   // [2, NE] row-major: src then dst
    float* out = (float*)d_out;

    // workspace layout (floats): deg/dinv | h1 | agg1(h) | h2
    float* ws   = (float*)d_ws;
    float* deg  = ws;                         // NN
    float* h1   = deg  + NN;                  // NN*HID
    float* agg1 = h1   + (size_t)NN * HID;    // NN*HID
    float* h2   = agg1 + (size_t)NN * HID;    // NN*OUT_F

    const int B = 256;
    k_init   <<<(NN * HID + B - 1) / B, B, 0, stream>>>(deg, agg1, out);
    k_degree <<<(NE + B - 1) / B,        B, 0, stream>>>(ei + NE, deg);
    k_dinv   <<<(NN + B - 1) / B,        B, 0, stream>>>(deg);

    size_t smem1 = (size_t)(IN_F * HID + 128 * XPAD) * sizeof(float); // ~98KB
    k_gemm1  <<<(NN + 127) / 128,        B, smem1, stream>>>(x, W1, h1);

    k_scatter1<<<((size_t)NE * HID + B - 1) / B, B, 0, stream>>>(ei, deg, h1, agg1);
    k_finish1 <<<(NN * HID + B - 1) / B,  B, 0, stream>>>(deg, h1, b1, agg1);

    k_gemm2  <<<(NN + B - 1) / B,        B, 0, stream>>>(agg1, W2, h2);
    k_scatter2<<<(NE + B - 1) / B,       B, 0, stream>>>(ei, deg, h2, out);
    k_finish2 <<<(NN * OUT_F + B - 1) / B, B, 0, stream>>>(deg, h2, b2, out);
}